// WeightedOhemCELoss_75084618269176
// MI455X (gfx1250) — compile-verified
//
#include <hip/hip_runtime.h>
#include <hip/hip_bf16.h>

// ---------------- problem constants (shapes fixed by reference) ----------------
#define CLS      19
#define HWSZ     262144          // 512*512
#define NPIX     4194304         // 16*512*512
#define NMIN     262144
#define THRESHF  0.35667494393873245f
#define G3       4096            // blocks for streaming passes (4096*256*4 == NPIX)
#define TPB      256

// ws layout (bytes):
//   0      : u32 hist1[4096]     (top-12-bit radix histogram)
//   16384  : u32 hist2[4096]     (next-12-bit histogram inside cutoff bucket)
//   32768  : u32 lhist[32]       (label histogram)
//   32896  : u32 misc[32]        [1]=cnt_gt [2]=Bk [3]=r [4]=cut24 [5]=t_take
//   33024  : f32 weights[32]
//   36864  : f32 partial_gt[4096]
//   53248  : f32 partial_ab[4096]
//   69632  : f32 partial_mid[4096]
//   131072 : f32 loss[NPIX]      (16.8 MB)

typedef __attribute__((ext_vector_type(2))) float v2f;
typedef __attribute__((ext_vector_type(8))) float v8f;

__device__ __forceinline__ float wave_reduce16(float v) {
  v += __shfl_xor(v, 1, 32);
  v += __shfl_xor(v, 2, 32);
  v += __shfl_xor(v, 4, 32);
  v += __shfl_xor(v, 8, 32);
  return v;
}

// Sum 256 fp32 values with chained V_WMMA_F32_16X16X4_F32 (A = ones).
// D[i][j] = sum_k B[k][j]; summing D row-values over 16 columns gives the total,
// independent of the exact (k,j) lane mapping. Must be called by a full wave.
__device__ __forceinline__ float wmma_reduce_256(const float* vals, int lane) {
#if defined(__gfx1250__) && __has_builtin(__builtin_amdgcn_wmma_f32_16x16x4_f32)
  v2f a; a.x = 1.0f; a.y = 1.0f;
  v8f acc = {};
  int col = lane & 15;
  int half = lane >> 4;
#pragma unroll
  for (int chunk = 0; chunk < 4; ++chunk) {
    const float* base = vals + chunk * 64;
    v2f b;
    b.x = base[half * 32 + col];
    b.y = base[half * 32 + col + 16];
    acc = __builtin_amdgcn_wmma_f32_16x16x4_f32(
        /*neg_a=*/false, a, /*neg_b=*/false, b,
        /*c_mod=*/(short)0, acc, /*reuse_a=*/false, /*reuse_b=*/false);
  }
  return wave_reduce16(acc[0]);   // every lane ends with the full 256-sum
#else
  float s = 0.f;
#pragma unroll
  for (int i = 0; i < 8; ++i) s += vals[lane * 8 + i];
  s += __shfl_xor(s, 16, 32);
  return wave_reduce16(s);
#endif
}

// ---------------- pass 1: label histogram ----------------
__global__ __launch_bounds__(256) void k_labhist(const int* __restrict__ labels,
                                                 unsigned* __restrict__ lhist) {
  __shared__ unsigned lh[CLS];
  int t = threadIdx.x;
  if (t < CLS) lh[t] = 0;
  __syncthreads();
  for (size_t i = (size_t)blockIdx.x * TPB + t; i < (size_t)NPIX;
       i += (size_t)gridDim.x * TPB) {
    int l = labels[i];
    if (l >= 0 && l < CLS) atomicAdd(&lh[l], 1u);
  }
  __syncthreads();
  if (t < CLS) { unsigned v = lh[t]; if (v) atomicAdd(&lhist[t], v); }
}

// ---------------- pass 2: ENet weights ----------------
__global__ void k_weights(const unsigned* __restrict__ lhist, float* __restrict__ w) {
  __shared__ float tot;
  int t = threadIdx.x;
  if (t == 0) {
    unsigned s = 0;
    for (int c = 0; c < CLS; ++c) s += lhist[c];
    tot = fmaxf((float)s, 1.0f);
  }
  __syncthreads();
  if (t < CLS) w[t] = 1.0f / logf(1.02f + (float)lhist[t] / tot);
}

// ---------------- pass 3: fused loss + radix-L1 histogram + thresh stats ----------------
__global__ __launch_bounds__(256) void k_loss(const float* __restrict__ logits,
                                              const int* __restrict__ labels,
                                              const float* __restrict__ w,
                                              float* __restrict__ loss,
                                              unsigned* __restrict__ hist1,
                                              float* __restrict__ partial_gt,
                                              unsigned* __restrict__ misc) {
  __shared__ float sw[CLS];
  __shared__ unsigned lh[4096];
  __shared__ float sbuf[TPB];
  __shared__ unsigned scnt;
  int t = threadIdx.x;
  if (t < CLS) sw[t] = w[t];
  if (t == 0) scnt = 0;
  for (int i = t; i < 4096; i += TPB) lh[i] = 0;
  __syncthreads();

  size_t p0 = ((size_t)blockIdx.x * TPB + t) * 4;  // 4 consecutive pixels, same image
  size_t b  = p0 / HWSZ;
  size_t hw = p0 - b * HWSZ;
  const float4* base = (const float4*)(logits + (b * CLS) * (size_t)HWSZ + hw);
  __builtin_prefetch((const char*)base + 65536, 0, 0);  // global_prefetch_b8

  int4 lab = *(const int4*)(labels + p0);
  int li[4] = {lab.x, lab.y, lab.z, lab.w};
  bool valid[4]; int safe[4];
#pragma unroll
  for (int j = 0; j < 4; ++j) {
    valid[j] = (li[j] != 255);
    safe[j]  = (valid[j] && li[j] >= 0 && li[j] < CLS) ? li[j] : 0;
  }

  float xs[CLS][4];                 // constant-indexed under full unroll -> registers
#pragma unroll
  for (int c = 0; c < CLS; ++c) {
    float4 v = base[(size_t)c * (HWSZ / 4)];   // coalesced b128, stride = class plane
    xs[c][0] = v.x; xs[c][1] = v.y; xs[c][2] = v.z; xs[c][3] = v.w;
  }

  float lossv[4];
  unsigned cgt = 0; float sgt = 0.f;
#pragma unroll
  for (int j = 0; j < 4; ++j) {
    float m = xs[0][j];
#pragma unroll
    for (int c = 1; c < CLS; ++c) m = fmaxf(m, xs[c][j]);
    float s = 0.f;
#pragma unroll
    for (int c = 0; c < CLS; ++c) s += __expf(xs[c][j] - m);
    float lse = m + __logf(s);
    float xl = xs[0][j];
#pragma unroll
    for (int c = 1; c < CLS; ++c) xl = (safe[j] == c) ? xs[c][j] : xl;
    float nll = lse - xl;
    float lv = valid[j] ? sw[safe[j]] * nll : 0.f;
    lv = fmaxf(lv, 0.f);            // keep nonneg so float bits are order-isomorphic
    lossv[j] = lv;
    atomicAdd(&lh[__float_as_uint(lv) >> 20], 1u);
    if (lv > THRESHF) { cgt++; sgt += lv; }
  }
  float4 outv = {lossv[0], lossv[1], lossv[2], lossv[3]};
  *(float4*)(loss + p0) = outv;

  atomicAdd(&scnt, cgt);
  sbuf[t] = sgt;
  __syncthreads();
  for (int i = t; i < 4096; i += TPB) { unsigned v = lh[i]; if (v) atomicAdd(&hist1[i], v); }
  for (int s = 128; s > 0; s >>= 1) {
    if (t < s) sbuf[t] += sbuf[t + s];
    __syncthreads();
  }
  if (t == 0) { partial_gt[blockIdx.x] = sbuf[0]; atomicAdd(&misc[1], scnt); }
}

// ---------------- radix select over a 4096-bin histogram (descending) ----------------
__global__ __launch_bounds__(256) void k_select(const unsigned* __restrict__ hist,
                                                unsigned* __restrict__ misc, int level) {
  __shared__ unsigned ssum[TPB];
  __shared__ unsigned sbefore[TPB];
  int t = threadIdx.x;
  unsigned target = (level == 1) ? (unsigned)NMIN : misc[3];
  int base = 4096 - 16 * (t + 1);           // chunk t = bins [base, base+15], top-down
  unsigned s = 0;
  for (int i = 0; i < 16; ++i) s += hist[base + i];
  ssum[t] = s;
  __syncthreads();
  if (t == 0) {
    unsigned run = 0;
    for (int i = 0; i < TPB; ++i) { sbefore[i] = run; run += ssum[i]; }
  }
  __syncthreads();
  unsigned before = sbefore[t];
  if (before < target && before + ssum[t] >= target) {   // unique owner
    unsigned cum = before;
    for (int i = 15; i >= 0; --i) {
      int bin = base + i;
      unsigned h = hist[bin];
      if (cum + h >= target) {
        unsigned take = target - cum;
        if (level == 1) { misc[2] = (unsigned)bin; misc[3] = take; }
        else { misc[4] = (misc[2] << 12) | (unsigned)bin; misc[5] = take; }
        break;
      }
      cum += h;
    }
  }
}

// ---------------- pass 5: sum-above-bucket + level-2 histogram ----------------
__global__ __launch_bounds__(256) void k_refine(const float* __restrict__ loss,
                                                const unsigned* __restrict__ misc,
                                                unsigned* __restrict__ hist2,
                                                float* __restrict__ partial_ab) {
  __shared__ unsigned lh[4096];
  __shared__ float sbuf[TPB];
  int t = threadIdx.x;
  for (int i = t; i < 4096; i += TPB) lh[i] = 0;
  __syncthreads();
  unsigned Bk = misc[2];
  size_t p0 = ((size_t)blockIdx.x * TPB + t) * 4;
  float4 lv = *(const float4*)(loss + p0);
  float vals[4] = {lv.x, lv.y, lv.z, lv.w};
  float sab = 0.f;
#pragma unroll
  for (int j = 0; j < 4; ++j) {
    unsigned bits = __float_as_uint(vals[j]);
    unsigned up = bits >> 20;
    if (up > Bk) sab += vals[j];
    else if (up == Bk) atomicAdd(&lh[(bits >> 8) & 0xFFFu], 1u);
  }
  sbuf[t] = sab;
  __syncthreads();
  for (int i = t; i < 4096; i += TPB) { unsigned v = lh[i]; if (v) atomicAdd(&hist2[i], v); }
  for (int s = 128; s > 0; s >>= 1) {
    if (t < s) sbuf[t] += sbuf[t + s];
    __syncthreads();
  }
  if (t == 0) partial_ab[blockIdx.x] = sbuf[0];
}

// ---------------- pass 7: sum of mid-band (inside L1 bucket, above L2 cut) ----------------
__global__ __launch_bounds__(256) void k_summid(const float* __restrict__ loss,
                                                const unsigned* __restrict__ misc,
                                                float* __restrict__ partial_mid) {
  __shared__ float sbuf[TPB];
  int t = threadIdx.x;
  unsigned Bk = misc[2];
  unsigned low = misc[4] & 0xFFFu;
  size_t p0 = ((size_t)blockIdx.x * TPB + t) * 4;
  float4 lv = *(const float4*)(loss + p0);
  float vals[4] = {lv.x, lv.y, lv.z, lv.w};
  float sm = 0.f;
#pragma unroll
  for (int j = 0; j < 4; ++j) {
    unsigned bits = __float_as_uint(vals[j]);
    unsigned up = bits >> 20;
    unsigned mid = (bits >> 8) & 0xFFFu;
    if (up == Bk && mid > low) sm += vals[j];
  }
  sbuf[t] = sm;
  __syncthreads();
  for (int s = 128; s > 0; s >>= 1) {
    if (t < s) sbuf[t] += sbuf[t + s];
    __syncthreads();
  }
  if (t == 0) partial_mid[blockIdx.x] = sbuf[0];
}

// ---------------- pass 8: final reduce (WMMA) + select answer ----------------
__global__ __launch_bounds__(256) void k_final(const float* __restrict__ pgt,
                                               const float* __restrict__ pab,
                                               const float* __restrict__ pmid,
                                               const unsigned* __restrict__ misc,
                                               float* __restrict__ out) {
  __shared__ float l256[TPB];
  __shared__ float res[3];
  int t = threadIdx.x;
  const float* arrs[3] = {pgt, pab, pmid};
#pragma unroll
  for (int a = 0; a < 3; ++a) {
    float s = 0.f;
#pragma unroll
    for (int i = 0; i < 16; ++i) s += arrs[a][t * 16 + i];   // fixed order
    l256[t] = s;
    __syncthreads();
    if (t < 32) {                         // wave 0, EXEC all-1s for WMMA
      float tot = wmma_reduce_256(l256, t);
      if (t == 0) res[a] = tot;
    }
    __syncthreads();
  }
  if (t == 0) {
    unsigned cnt_gt = misc[1];
    float mean_gt = res[0] / fmaxf((float)cnt_gt, 1.0f);
    unsigned cut24 = misc[4];
    float vstar = __uint_as_float((cut24 << 8) | 0x80u);     // tie representative
    float sum_top = res[1] + res[2] + (float)misc[5] * vstar;
    float mean_top = sum_top / (float)NMIN;
    // count(loss > THRESH) > NMIN  <=>  s[NMIN] > THRESH
    out[0] = (cnt_gt > (unsigned)NMIN) ? mean_gt : mean_top;
  }
}

extern "C" void kernel_launch(void* const* d_in, const int* in_sizes, int n_in,
                              void* d_out, int out_size, void* d_ws, size_t ws_size,
                              hipStream_t stream) {
  const float* logits = (const float*)d_in[0];
  const int*   labels = (const int*)d_in[1];
  char* ws = (char*)d_ws;
  unsigned* hist1 = (unsigned*)(ws + 0);
  unsigned* hist2 = (unsigned*)(ws + 16384);
  unsigned* lhist = (unsigned*)(ws + 32768);
  unsigned* misc  = (unsigned*)(ws + 32896);
  float*    wts   = (float*)   (ws + 33024);
  float*    pgt   = (float*)   (ws + 36864);
  float*    pab   = (float*)   (ws + 53248);
  float*    pmid  = (float*)   (ws + 69632);
  float*    loss  = (float*)   (ws + 131072);

  hipMemsetAsync(d_ws, 0, 36864, stream);                    // zero hists + misc
  k_labhist<<<2048, TPB, 0, stream>>>(labels, lhist);
  k_weights<<<1, 32, 0, stream>>>(lhist, wts);
  k_loss<<<G3, TPB, 0, stream>>>(logits, labels, wts, loss, hist1, pgt, misc);
  k_select<<<1, TPB, 0, stream>>>(hist1, misc, 1);
  k_refine<<<G3, TPB, 0, stream>>>(loss, misc, hist2, pab);
  k_select<<<1, TPB, 0, stream>>>(hist2, misc, 2);
  k_summid<<<G3, TPB, 0, stream>>>(loss, misc, pmid);
  k_final<<<1, TPB, 0, stream>>>(pgt, pab, pmid, misc, (float*)d_out);
}